// LengthRegulator_52742198395187
// MI455X (gfx1250) — compile-verified
//
#include <hip/hip_runtime.h>
#include <hip/hip_bf16.h>

#define BATCH   32
#define LSEQ    1024
#define DMODEL  384
#define ROWBYTES (DMODEL * 4)         // 1536 B per row
#define WAVES_PER_BLOCK 8             // 256 threads, wave32 -> 8 rows per block

typedef float v4f __attribute__((ext_vector_type(4)));

// Truncate a generic (flat) pointer to the 32-bit LDS byte offset.
// Per CDNA5 ISA 10.2, the LDS aperture keeps the wave-relative LDS address
// in addr[31:0]; async-to-LDS instructions take exactly this offset.
__device__ __forceinline__ uint32_t lds_addr(const void* p) {
    return (uint32_t)(uintptr_t)p;
}

// ---------------------------------------------------------------------------
// Kernel 1: per-batch inclusive scan of round(clip(durations,0)).
// One block of LSEQ threads per batch; Hillis-Steele in LDS (10 steps).
// ---------------------------------------------------------------------------
__global__ void lr_scan_kernel(const float* __restrict__ dur,
                               int* __restrict__ cum) {
    __shared__ int s[LSEQ];
    const int b   = blockIdx.x;
    const int tid = threadIdx.x;

    float d = dur[b * LSEQ + tid];
    d = fmaxf(d, 0.0f);
    s[tid] = (int)rintf(d);           // RNE == jnp.round (banker's rounding)
    __syncthreads();

#pragma unroll
    for (int off = 1; off < LSEQ; off <<= 1) {
        int add = (tid >= off) ? s[tid - off] : 0;
        __syncthreads();
        s[tid] += add;
        __syncthreads();
    }
    cum[b * LSEQ + tid] = s[tid];
}

// ---------------------------------------------------------------------------
// Kernel 2: one wave32 per output row, data path entirely through the CDNA5
// async global<->LDS engine (ASYNCcnt), never touching VGPRs:
//   global_load_async_to_lds_b128    (gather x row -> LDS stage, RT: x is
//                                     50 MB with ~6x reuse, keep in L2)
//   global_store_async_from_lds_b128 th:TH_STORE_NT (stream 403 MB out
//                                     without polluting L2)
// ---------------------------------------------------------------------------
__global__ void lr_gather_kernel(const float* __restrict__ x,
                                 const int* __restrict__ cum,
                                 float* __restrict__ out,
                                 int max_len) {
    __shared__ __align__(16) int   scum[LSEQ];                       // 4 KB
    __shared__ __align__(16) float stage[WAVES_PER_BLOCK * DMODEL];  // 12 KB

    const int b    = blockIdx.y;
    const int tid  = threadIdx.x;
    const int wave = tid >> 5;
    const int lane = tid & 31;

    // ---- async-stage cum[b][0:LSEQ] into LDS: 256 threads x 16 B = 4 KB ----
    {
        uint32_t l = lds_addr(scum) + (uint32_t)tid * 16u;
        uint32_t g = ((uint32_t)b * LSEQ + (uint32_t)tid * 4u) * 4u;
        asm volatile("global_load_async_to_lds_b128 %0, %1, %2"
                     :: "v"(l), "v"(g), "s"(cum) : "memory");
        asm volatile("s_wait_asynccnt 0x0" ::: "memory");
    }
    __syncthreads();

    const int t = blockIdx.x * WAVES_PER_BLOCK + wave;
    if (t >= max_len) return;

    const int total = scum[LSEQ - 1];
    const uint32_t mystage = lds_addr(stage) + (uint32_t)wave * ROWBYTES;
    const uint32_t lbase   = mystage + (uint32_t)lane * 16u;

    if (t < total) {
        // upper_bound: first i with scum[i] > t  (searchsorted side='right')
        int lo = 0, hi = LSEQ;
        while (lo < hi) {
            int mid = (lo + hi) >> 1;
            if (scum[mid] <= t) lo = mid + 1; else hi = mid;
        }
        const int idx = lo < (LSEQ - 1) ? lo : (LSEQ - 1);
        const uint32_t gbase =
            ((uint32_t)b * LSEQ + (uint32_t)idx) * (uint32_t)ROWBYTES
            + (uint32_t)lane * 16u;
        // 3 x (32 lanes x 16 B) = 1536 B row, gathered into this wave's slice
        asm volatile("global_load_async_to_lds_b128 %0, %1, %2"
                     :: "v"(lbase),         "v"(gbase),         "s"(x) : "memory");
        asm volatile("global_load_async_to_lds_b128 %0, %1, %2"
                     :: "v"(lbase + 512u),  "v"(gbase + 512u),  "s"(x) : "memory");
        asm volatile("global_load_async_to_lds_b128 %0, %1, %2"
                     :: "v"(lbase + 1024u), "v"(gbase + 1024u), "s"(x) : "memory");
    } else {
        // masked row: fill stage slice with zeros (ds_store_b128)
        v4f z = (v4f)0.0f;
        v4f* sp = (v4f*)(stage + wave * DMODEL);
        sp[lane]      = z;
        sp[lane + 32] = z;
        sp[lane + 64] = z;
    }

    // Async stores read LDS out-of-order vs async loads / DS ops:
    // drain both counters before streaming the row out.
    asm volatile("s_wait_asynccnt 0x0" ::: "memory");
    asm volatile("s_wait_dscnt 0x0"    ::: "memory");

    const uint32_t obase =
        ((uint32_t)b * (uint32_t)max_len + (uint32_t)t) * (uint32_t)ROWBYTES
        + (uint32_t)lane * 16u;
    asm volatile("global_store_async_from_lds_b128 %0, %1, %2 th:TH_STORE_NT"
                 :: "v"(obase),         "v"(lbase),         "s"(out) : "memory");
    asm volatile("global_store_async_from_lds_b128 %0, %1, %2 th:TH_STORE_NT"
                 :: "v"(obase + 512u),  "v"(lbase + 512u),  "s"(out) : "memory");
    asm volatile("global_store_async_from_lds_b128 %0, %1, %2 th:TH_STORE_NT"
                 :: "v"(obase + 1024u), "v"(lbase + 1024u), "s"(out) : "memory");
    // S_ENDPGM performs an implicit wait-idle, draining outstanding async stores.
}

// ---------------------------------------------------------------------------
extern "C" void kernel_launch(void* const* d_in, const int* in_sizes, int n_in,
                              void* d_out, int out_size, void* d_ws, size_t ws_size,
                              hipStream_t stream) {
    const float* x   = (const float*)d_in[0];   // (B, L, D) f32
    const float* dur = (const float*)d_in[1];   // (B, L)    f32
    float* out = (float*)d_out;
    int*   cum = (int*)d_ws;                    // B*L ints = 128 KB scratch

    const int max_len = out_size / (BATCH * DMODEL);   // 8192

    lr_scan_kernel<<<dim3(BATCH), dim3(LSEQ), 0, stream>>>(dur, cum);

    dim3 grid((max_len + WAVES_PER_BLOCK - 1) / WAVES_PER_BLOCK, BATCH);
    lr_gather_kernel<<<grid, dim3(32 * WAVES_PER_BLOCK), 0, stream>>>(x, cum, out, max_len);
}